// NewFastRCNNOutputLayers_36120674959977
// MI455X (gfx1250) — compile-verified
//
#include <hip/hip_runtime.h>

#define B_ 8
#define R_ 2000
#define K_ 80
#define NFLAT (R_ * K_)
#define M_ 2048
#define T_ 100
#define W_IMG 1333.0f
#define H_IMG 800.0f
#define SCORE_TH 0.05f
#define NMS_TH 0.5f
#define CLS_SHIFT 1335.0f   /* max(H,W) + 2 */
#define NTHREADS 1024

typedef float v2f __attribute__((ext_vector_type(2)));
typedef float v8f __attribute__((ext_vector_type(8)));

__device__ __forceinline__ bool finitef(float x) {
  return (__float_as_uint(x) & 0x7F800000u) != 0x7F800000u;
}

// Constant 8x10 decode matrix W (padded to 16 cols), entries in {-1,0,1}:
// features: [cx, cy, b0*dx, b1*dy, b2*dx, b3*dy, b4*dx, b5*dy]
// outputs:  [px1 py1 px2 py2 px3 py3 px4 py4 midx midy]
__device__ __forceinline__ float wval(int k, int n) {
  switch (k) {
    case 0: case 2: return (n <= 8 && (n & 1) == 0) ? 1.0f : 0.0f; // cx / b0*dx
    case 1: case 3: return (n <= 9 && (n & 1) == 1) ? 1.0f : 0.0f; // cy / b1*dy
    case 4: return (n == 0) ? 1.0f : ((n == 6) ? -1.0f : 0.0f);    // b2*dx (firstx)
    case 5: return (n == 1) ? 1.0f : ((n == 7) ? -1.0f : 0.0f);    // b3*dy (firsty)
    case 6: return (n == 2) ? 1.0f : ((n == 4) ? -1.0f : 0.0f);    // b4*dx (secondx)
    case 7: return (n == 3) ? 1.0f : ((n == 5) ? -1.0f : 0.0f);    // b5*dy (secondy)
  }
  return 0.0f;
}

__global__ __launch_bounds__(NTHREADS)
void rcnn_post_kernel(const float* __restrict__ boxes,
                      const float* __restrict__ scores,
                      const float* __restrict__ bases,
                      float* __restrict__ out) {
  const int b = blockIdx.x;
  const int tid = (int)threadIdx.x;

  __shared__ float s_score[M_];        // 8 KB  candidate scores (sorted desc)
  __shared__ unsigned s_idx[M_];       // 8 KB  flat index r*K+c
  __shared__ float s_x1[M_];           // 32 KB offset boxes for IoU
  __shared__ float s_y1[M_];
  __shared__ float s_x2[M_];
  __shared__ float s_y2[M_];
  __shared__ float s_area[M_];         // 8 KB
  __shared__ unsigned char s_supp[M_]; // 2 KB
  __shared__ unsigned char s_keep[M_]; // 2 KB
  __shared__ int s_selT[T_];
  __shared__ unsigned sg_b1, sg_cab, sg_b2, sg_cab2;
  __shared__ unsigned sg_thr, sg_cg, sg_teq, sg_ki, sg_posG, sg_posE, sg_skip;
  __shared__ int sg_nkeep;

  unsigned* hist = (unsigned*)s_x1;       // alias: used only in radix passes
  unsigned char* sFin = s_supp;           // alias: used only before NMS
  float* sF = s_area;                     // alias: 112*8 features, post-NMS

  const float* bx = boxes + (size_t)b * R_ * 4;
  const float* sc = scores + (size_t)b * R_ * (K_ + 1);
  const float* bs = bases + (size_t)b * R_ * 7;
  const float NEG_INF = __int_as_float(0xFF800000);

  // ---------- A0: per-row finite mask (boxes row + all 81 score cols) ----------
  for (int r = tid; r < R_; r += NTHREADS) {
    bool f = true;
#pragma unroll
    for (int j = 0; j < 4; ++j) f = f && finitef(bx[r * 4 + j]);
    for (int j = 0; j <= K_; ++j) f = f && finitef(sc[r * (K_ + 1) + j]);
    sFin[r] = f ? 1 : 0;
  }
  __syncthreads();

  // ---------- A1: radix pass 1 (key bits [31:21], 2048 buckets) ----------
  for (int i = tid; i < 2048; i += NTHREADS) hist[i] = 0;
  __syncthreads();
  for (int t = tid; t < NFLAT; t += NTHREADS) {
    int r = t / K_; int c = t - r * K_;
    float s = sc[r * (K_ + 1) + c];
    if (sFin[r] && s > SCORE_TH)
      atomicAdd(&hist[__float_as_uint(s) >> 21], 1u);
  }
  __syncthreads();
  if (tid == 0) {
    unsigned cum = 0, b1 = 0, cab = 0; int found = 0;
    for (int bk = 2047; bk >= 0; --bk) {
      unsigned h = hist[bk];
      if (!found && cum + h >= (unsigned)M_) { b1 = (unsigned)bk; cab = cum; found = 1; }
      cum += h;
    }
    if (!found) { sg_skip = 1u; sg_thr = 0u; sg_cg = 0u; sg_teq = 0u; }
    else        { sg_skip = 0u; sg_b1 = b1; sg_cab = cab; }
  }
  __syncthreads();

  if (!sg_skip) {
    // ---------- A2: radix pass 2 (bits [20:10] within boundary bucket) ----------
    for (int i = tid; i < 2048; i += NTHREADS) hist[i] = 0;
    __syncthreads();
    {
      unsigned b1 = sg_b1;
      for (int t = tid; t < NFLAT; t += NTHREADS) {
        int r = t / K_; int c = t - r * K_;
        float s = sc[r * (K_ + 1) + c];
        if (sFin[r] && s > SCORE_TH) {
          unsigned kk = __float_as_uint(s);
          if ((kk >> 21) == b1) atomicAdd(&hist[(kk >> 10) & 0x7FFu], 1u);
        }
      }
    }
    __syncthreads();
    if (tid == 0) {
      unsigned cum = sg_cab;
      for (int bk = 2047; bk >= 0; --bk) {
        unsigned h = hist[bk];
        if (cum + h >= (unsigned)M_) { sg_b2 = (sg_b1 << 11) | (unsigned)bk; sg_cab2 = cum; break; }
        cum += h;
      }
    }
    __syncthreads();
    // ---------- A3: radix pass 3 (bits [9:0] -> exact threshold key) ----------
    for (int i = tid; i < 2048; i += NTHREADS) hist[i] = 0;
    __syncthreads();
    {
      unsigned p22 = sg_b2;
      for (int t = tid; t < NFLAT; t += NTHREADS) {
        int r = t / K_; int c = t - r * K_;
        float s = sc[r * (K_ + 1) + c];
        if (sFin[r] && s > SCORE_TH) {
          unsigned kk = __float_as_uint(s);
          if ((kk >> 10) == p22) atomicAdd(&hist[kk & 0x3FFu], 1u);
        }
      }
    }
    __syncthreads();
    if (tid == 0) {
      unsigned cum = sg_cab2;
      for (int bk = 1023; bk >= 0; --bk) {
        unsigned h = hist[bk];
        if (cum + h >= (unsigned)M_) {
          sg_thr = (sg_b2 << 10) | (unsigned)bk;
          sg_cg = cum;
          sg_teq = (unsigned)M_ - cum;
          break;
        }
        cum += h;
      }
    }
    __syncthreads();
  }

  // ---------- A4: compaction of the exact top-M candidates ----------
  for (int i = tid; i < M_; i += NTHREADS) { s_score[i] = NEG_INF; s_idx[i] = 0u; }
  if (tid == 0) { sg_posG = 0u; sg_posE = sg_cg; }
  __syncthreads();
  {
    unsigned thr = sg_thr, lim = sg_cg + sg_teq;
    for (int t = tid; t < NFLAT; t += NTHREADS) {
      int r = t / K_; int c = t - r * K_;
      float s = sc[r * (K_ + 1) + c];
      if (!(sFin[r] && s > SCORE_TH)) continue;
      unsigned kk = __float_as_uint(s);
      if (kk > thr) {
        unsigned p = atomicAdd(&sg_posG, 1u);
        if (p < (unsigned)M_) { s_score[p] = s; s_idx[p] = (unsigned)t; }
      } else if (kk == thr) {
        unsigned p = atomicAdd(&sg_posE, 1u);
        if (p < lim) { s_score[p] = s; s_idx[p] = (unsigned)t; }
      }
    }
  }
  __syncthreads();

  // ---------- A5: bitonic sort 2048 in LDS (descending score, ties by idx) ----------
  for (unsigned k = 2; k <= (unsigned)M_; k <<= 1) {
    for (unsigned j = k >> 1; j > 0; j >>= 1) {
      for (unsigned i = (unsigned)tid; i < (unsigned)M_; i += NTHREADS) {
        unsigned l = i ^ j;
        if (l > i) {
          float si = s_score[i], sl = s_score[l];
          unsigned ii = s_idx[i], il = s_idx[l];
          bool dir = ((i & k) == 0);
          bool lBefore = (sl > si) || (sl == si && il < ii);
          bool iBefore = (si > sl) || (si == sl && ii < il);
          bool sw = dir ? lBefore : iBefore;
          if (sw) { s_score[i] = sl; s_score[l] = si; s_idx[i] = il; s_idx[l] = ii; }
        }
      }
      __syncthreads();
    }
  }

  // ---------- A6: gather clipped + class-offset boxes into LDS ----------
  for (int i = tid; i < M_; i += NTHREADS) {
    float scv = s_score[i];
    if (scv != NEG_INF) {
      unsigned flat = s_idx[i];
      int r = (int)(flat / K_); int c = (int)(flat - (unsigned)r * K_);
      float x1 = fminf(fmaxf(bx[r * 4 + 0], 0.f), W_IMG);
      float y1 = fminf(fmaxf(bx[r * 4 + 1], 0.f), H_IMG);
      float x2 = fminf(fmaxf(bx[r * 4 + 2], 0.f), W_IMG);
      float y2 = fminf(fmaxf(bx[r * 4 + 3], 0.f), H_IMG);
      float off = (float)c * CLS_SHIFT;
      x1 += off; y1 += off; x2 += off; y2 += off;
      s_x1[i] = x1; s_y1[i] = y1; s_x2[i] = x2; s_y2[i] = y2;
      s_area[i] = fmaxf(x2 - x1, 0.f) * fmaxf(y2 - y1, 0.f);
    } else {
      s_x1[i] = 0.f; s_y1[i] = 0.f; s_x2[i] = 0.f; s_y2[i] = 0.f; s_area[i] = 0.f;
    }
  }
  __syncthreads();
  for (int i = tid; i < M_; i += NTHREADS) { s_supp[i] = 0; s_keep[i] = 0; }
  __syncthreads();

  // ---------- B: sequential greedy NMS, IoU rows on the fly (LDS resident) ----------
  for (int i = 0; i < M_; ++i) {
    if (tid == 0) {
      unsigned ki = (s_score[i] != NEG_INF && !s_supp[i]) ? 1u : 0u;
      s_keep[i] = (unsigned char)ki;
      sg_ki = ki;
    }
    __syncthreads();
    if (sg_ki) {
      float x1i = s_x1[i], y1i = s_y1[i], x2i = s_x2[i], y2i = s_y2[i], ai = s_area[i];
      for (int j = tid; j < M_; j += NTHREADS) {
        float xx1 = fmaxf(x1i, s_x1[j]);
        float yy1 = fmaxf(y1i, s_y1[j]);
        float xx2 = fminf(x2i, s_x2[j]);
        float yy2 = fminf(y2i, s_y2[j]);
        float inter = fmaxf(xx2 - xx1, 0.f) * fmaxf(yy2 - yy1, 0.f);
        float iou = inter / (ai + s_area[j] - inter + 1e-7f);
        if (iou > NMS_TH) s_supp[j] = 1;
      }
    }
    __syncthreads();
  }

  // ---------- C: first T kept (already score-descending) ----------
  if (tid == 0) {
    int cnt = 0;
    for (int i = 0; i < M_ && cnt < T_; ++i)
      if (s_keep[i]) s_selT[cnt++] = i;
    sg_nkeep = cnt;
  }
  __syncthreads();

  float* out_dec = out;
  float* out_h   = out + B_ * T_ * 10;
  float* out_bx  = out_h + B_ * T_;
  float* out_sc  = out_bx + B_ * T_ * 4;
  float* out_cls = out_sc + B_ * T_;
  float* out_val = out_cls + B_ * T_;

  // ---------- D1: per-survivor features + scalar outputs (rows padded to 112) ----------
  int nk = sg_nkeep;
  if (tid < 112) {
    int t = tid;
    float f[8] = {0.f, 0.f, 0.f, 0.f, 0.f, 0.f, 0.f, 0.f};
    if (t < T_) {
      if (t < nk) {
        int i = s_selT[t];
        unsigned flat = s_idx[i];
        int r = (int)(flat / K_); int c = (int)(flat - (unsigned)r * K_);
        float x1 = fminf(fmaxf(bx[r * 4 + 0], 0.f), W_IMG);
        float y1 = fminf(fmaxf(bx[r * 4 + 1], 0.f), H_IMG);
        float x2 = fminf(fmaxf(bx[r * 4 + 2], 0.f), W_IMG);
        float y2 = fminf(fmaxf(bx[r * 4 + 3], 0.f), H_IMG);
        float dx = x2 - x1, dy = y2 - y1;
        const float* pb = bs + (size_t)r * 7;
        f[0] = (x1 + x2) * 0.5f;  f[1] = (y1 + y2) * 0.5f;
        f[2] = pb[0] * dx;        f[3] = pb[1] * dy;
        f[4] = pb[2] * dx;        f[5] = pb[3] * dy;
        f[6] = pb[4] * dx;        f[7] = pb[5] * dy;
        out_h[b * T_ + t] = dy + pb[6] * dy;
        out_bx[(b * T_ + t) * 4 + 0] = x1;
        out_bx[(b * T_ + t) * 4 + 1] = y1;
        out_bx[(b * T_ + t) * 4 + 2] = x2;
        out_bx[(b * T_ + t) * 4 + 3] = y2;
        out_sc[b * T_ + t]  = s_score[i];
        out_cls[b * T_ + t] = (float)c;
        out_val[b * T_ + t] = 1.0f;
      } else {
        out_h[b * T_ + t] = 0.f;
        out_bx[(b * T_ + t) * 4 + 0] = 0.f;
        out_bx[(b * T_ + t) * 4 + 1] = 0.f;
        out_bx[(b * T_ + t) * 4 + 2] = 0.f;
        out_bx[(b * T_ + t) * 4 + 3] = 0.f;
        out_sc[b * T_ + t]  = 0.f;
        out_cls[b * T_ + t] = -1.0f;
        out_val[b * T_ + t] = 0.f;
      }
    }
#pragma unroll
    for (int kk = 0; kk < 8; ++kk) sF[t * 8 + kk] = f[kk];
  }
  __syncthreads();

  // ---------- D2: dec[100,10] = F[100,8] @ W[8,10] via V_WMMA_F32_16X16X4_F32 ----------
  // Waves 0..6 each produce one 16-row tile; K=8 done as two K=4 WMMA steps.
  {
    int wid = tid >> 5;
    int lane = tid & 31;
    if (wid < 7) {                      // wave-uniform: EXEC all ones inside
      int base = wid * 16;
      int mrow = lane & 15;
      int khalf = lane >> 4;            // A layout: lanes 0-15 hold K0/K1, 16-31 hold K2/K3
      v2f aLo, aHi, bLo, bHi;
      aLo.x = sF[(base + mrow) * 8 + khalf * 2 + 0];
      aLo.y = sF[(base + mrow) * 8 + khalf * 2 + 1];
      aHi.x = sF[(base + mrow) * 8 + 4 + khalf * 2 + 0];
      aHi.y = sF[(base + mrow) * 8 + 4 + khalf * 2 + 1];
      int n = lane & 15;                // B layout mirror: lane = column
      bLo.x = wval(khalf * 2 + 0, n);
      bLo.y = wval(khalf * 2 + 1, n);
      bHi.x = wval(4 + khalf * 2 + 0, n);
      bHi.y = wval(4 + khalf * 2 + 1, n);
      v8f acc = {0.f, 0.f, 0.f, 0.f, 0.f, 0.f, 0.f, 0.f};
      acc = __builtin_amdgcn_wmma_f32_16x16x4_f32(false, aLo, false, bLo,
                                                  (short)0, acc, false, false);
      acc = __builtin_amdgcn_wmma_f32_16x16x4_f32(false, aHi, false, bHi,
                                                  (short)0, acc, false, false);
      union { v8f v; float f[8]; } du; du.v = acc;
#pragma unroll
      for (int v = 0; v < 8; ++v) {
        int row = base + (khalf ? v + 8 : v);   // C/D: lanes 0-15 -> M=v, 16-31 -> M=v+8
        if (row < T_ && n < 10)
          out_dec[(b * T_ + row) * 10 + n] = du.f[v];
      }
    }
  }
}

extern "C" void kernel_launch(void* const* d_in, const int* in_sizes, int n_in,
                              void* d_out, int out_size, void* d_ws, size_t ws_size,
                              hipStream_t stream) {
  (void)in_sizes; (void)n_in; (void)out_size; (void)d_ws; (void)ws_size;
  const float* boxes  = (const float*)d_in[0];
  const float* scores = (const float*)d_in[1];
  const float* bases  = (const float*)d_in[2];
  float* out = (float*)d_out;
  hipLaunchKernelGGL(rcnn_post_kernel, dim3(B_), dim3(NTHREADS), 0, stream,
                     boxes, scores, bases, out);
}